// LIF_41283225649587
// MI455X (gfx1250) — compile-verified
//
#include <hip/hip_runtime.h>
#include <hip/hip_bf16.h>

// ---------------------------------------------------------------------------
// LIF scan for MI455X (gfx1250).
// Memory-bound streaming recurrence (8192 independent rows x 4096 serial
// steps).  Parallelism is capped at 256 wave32s, so effective HBM bandwidth is
// set by bytes-in-flight per wave: use b128 async global->LDS DMA (ASYNCcnt)
// with a 4-buffer / 3-chunks-ahead pipeline = 24 ops x 512B = 12KB in flight
// per wave, vs 4KB with the previous b32 double-buffer.
// ---------------------------------------------------------------------------

typedef int v4i __attribute__((ext_vector_type(4)));

#define GLOBAL_AS __attribute__((address_space(1)))
#define LDS_AS    __attribute__((address_space(3)))

#if defined(__has_builtin)
#  if __has_builtin(__builtin_amdgcn_global_load_async_to_lds_b128)
#    define HAVE_ASYNC_LDS 1
#  endif
#  if __has_builtin(__builtin_amdgcn_s_wait_asynccnt)
#    define HAVE_WAIT_ASYNC 1
#  endif
#endif
#ifndef HAVE_ASYNC_LDS
#  define HAVE_ASYNC_LDS 0
#endif
#ifndef HAVE_WAIT_ASYNC
#  define HAVE_WAIT_ASYNC 0
#endif

#if HAVE_WAIT_ASYNC
#  define WAIT_ASYNC(n) __builtin_amdgcn_s_wait_asynccnt(n)
#elif HAVE_ASYNC_LDS
#  define WAIT_ASYNC(n) asm volatile("s_wait_asynccnt %0" ::"i"(n) : "memory")
#else
#  define WAIT_ASYNC(n) ((void)0)
#endif

namespace {
constexpr int   L_LEN          = 4096;   // timesteps per row (reference L)
constexpr int   ROWS_PER_BLOCK = 256;    // one thread == one row
constexpr int   TC             = 32;     // time chunk (128B per row-chunk)
constexpr int   NC             = L_LEN / TC;          // 128 chunks
constexpr int   NBUF           = 4;      // LDS pipeline depth
constexpr int   AHEAD          = 3;      // chunks prefetched ahead
constexpr int   OPS_PER_CHUNK  = TC / 4; // 8 b128 ops per thread per chunk
// Row-major LDS tile: lds[row * ROW_STRIDE + col].  Stride 36 floats keeps
// every b128 row 16B-aligned; compute reads see only a 2-way bank conflict
// (36*16 % 64 == 0), trivial against LDS bandwidth.
constexpr int   ROW_STRIDE     = TC + 4;              // 36 floats
constexpr int   BUF_FLOATS     = ROWS_PER_BLOCK * ROW_STRIDE;
constexpr float V_DECAY        = 1.0f - 1.0f / 20.0f; // 1 - 1/TAU
constexpr float V_TH           = 1.0f;
constexpr float K_SHARP        = 10.0f;
}  // namespace

// Issue the async (or fallback) b128 loads for one [256 rows x TC cols] tile.
// vec4 index q = k*256 + tid -> row = q/8, seg = q%8: lanes 0..7 cover one
// 128B-contiguous row chunk (coalesced), 4 rows per wave instruction.
__device__ __forceinline__ void issue_chunk_load(const float* __restrict__ I,
                                                 float* __restrict__ sbuf,
                                                 int r0, int t0, int tid) {
#pragma unroll
  for (int k = 0; k < OPS_PER_CHUNK; ++k) {
    const int q   = k * ROWS_PER_BLOCK + tid;
    const int row = q >> 3;        // q / (TC/4)
    const int seg = q & 7;         // q % (TC/4)
    const float* g = I + (size_t)(r0 + row) * L_LEN + (size_t)(t0 + seg * 4);
    float* l = sbuf + row * ROW_STRIDE + seg * 4;
#if HAVE_ASYNC_LDS
    __builtin_amdgcn_global_load_async_to_lds_b128(
        (GLOBAL_AS v4i*)g, (LDS_AS v4i*)l, /*offset=*/0, /*cpol=*/0);
#else
    *(float4*)l = *(const float4*)g;  // synchronous fallback, same layout
#endif
  }
}

__global__ void __launch_bounds__(ROWS_PER_BLOCK)
lif_scan_kernel(const float* __restrict__ I, float* __restrict__ spikes,
                long long* __restrict__ hard_lat, float* __restrict__ soft_lat) {
  __shared__ float sI[NBUF * BUF_FLOATS];  // 144 KB of the 320 KB WGP LDS

  const int tid = threadIdx.x;
  const int r0  = blockIdx.x * ROWS_PER_BLOCK;
  const int row = r0 + tid;

  // Prologue: fill the pipeline with chunks 0..AHEAD-1.
#pragma unroll
  for (int p = 0; p < AHEAD; ++p) {
    issue_chunk_load(I, sI + p * BUF_FLOATS, r0, p * TC, tid);
  }

  float v      = 0.0f;
  float acc_s  = 0.0f;   // sum of surrogate spikes
  float acc_st = 0.0f;   // sum of s * t
  int   first  = L_LEN;  // hard first-spike time (L if never)

  for (int c = 0; c < NC; ++c) {
    const int t0  = c * TC;
    float*    cur = sI + (c & (NBUF - 1)) * BUF_FLOATS;

    // Keep AHEAD chunks in flight; buffer (c+AHEAD)%NBUF was drained at
    // iteration c-1 (trailing barrier orders the reuse).
    if (c + AHEAD < NC) {
      issue_chunk_load(I, sI + ((c + AHEAD) & (NBUF - 1)) * BUF_FLOATS, r0,
                       t0 + AHEAD * TC, tid);
    }
    // Async loads complete in order: once <= 8*ahead newer ops remain, all of
    // chunk c's ops are done.  Tighten the threshold at the loop tail.
    const int ahead = (NC - 1 - c < AHEAD) ? (NC - 1 - c) : AHEAD;
    if (ahead == 3)      WAIT_ASYNC(3 * OPS_PER_CHUNK);
    else if (ahead == 2) WAIT_ASYNC(2 * OPS_PER_CHUNK);
    else if (ahead == 1) WAIT_ASYNC(1 * OPS_PER_CHUNK);
    else                 WAIT_ASYNC(0);
    __syncthreads();  // all waves' loads for chunk c are now visible

    const float* myrow = cur + tid * ROW_STRIDE;
    float sv[TC];
#pragma unroll
    for (int tl = 0; tl < TC; ++tl) {
      const float It = myrow[tl];
      v = fmaf(v, V_DECAY, It);                         // Euler integrate
      const float s = 1.0f / (1.0f + __expf(-K_SHARP * (v - V_TH)));
      if (first == L_LEN && v > V_TH) first = t0 + tl;  // hard spike
      sv[tl]  = s;
      acc_s  += s;
      acc_st += s * (float)(t0 + tl);
      v *= (1.0f - s);                                  // stopgrad reset
    }

    // 8 x b128 stores per thread: 128B per lane per chunk, line-fillable.
    float4* outp = (float4*)(spikes + (size_t)row * L_LEN + t0);
#pragma unroll
    for (int j = 0; j < TC / 4; ++j) {
      outp[j] = make_float4(sv[4 * j + 0], sv[4 * j + 1],
                            sv[4 * j + 2], sv[4 * j + 3]);
    }
    __syncthreads();  // buffer (c&3) may be refilled at iteration c+1
  }

  hard_lat[row] = (long long)first;
  soft_lat[row] = acc_st / (acc_s + 1e-6f);
}

extern "C" void kernel_launch(void* const* d_in, const int* in_sizes, int n_in,
                              void* d_out, int out_size, void* d_ws,
                              size_t ws_size, hipStream_t stream) {
  (void)n_in; (void)out_size; (void)d_ws; (void)ws_size;

  const float* I = (const float*)d_in[0];
  const int total = in_sizes[0];        // B * L
  const int B     = total / L_LEN;      // 8192

  // d_out = [spikes fp32 B*L][hard_latency int64 B][soft_latency fp32 B]
  char*      base   = (char*)d_out;
  float*     spikes = (float*)base;
  long long* hard   = (long long*)(base + (size_t)B * L_LEN * sizeof(float));
  float*     soft   = (float*)(base + (size_t)B * L_LEN * sizeof(float) +
                               (size_t)B * sizeof(long long));

  dim3 grid(B / ROWS_PER_BLOCK);
  dim3 block(ROWS_PER_BLOCK);
  lif_scan_kernel<<<grid, block, 0, stream>>>(I, spikes, hard, soft);
}